// Model_67293547594181
// MI455X (gfx1250) — compile-verified
//
#include <hip/hip_runtime.h>

// ---------------- Problem constants (from reference) ----------------
#define B_     8
#define H_     32
#define KVH_   8
#define D_     128
#define PS_    16
#define LMAX_  4096
#define WAVES_ 16   // waves per block (block = 512 threads)

typedef __attribute__((ext_vector_type(2))) float v2f;
typedef __attribute__((ext_vector_type(8))) float v8f;

// wave32 lane broadcast / exchange via LDS-permute hardware (no LDS memory used)
__device__ __forceinline__ float lanebcast_f(float v, int srcLane) {
    return __int_as_float(__builtin_amdgcn_ds_bpermute(srcLane << 2, __float_as_int(v)));
}
__device__ __forceinline__ int lanebcast_i(int v, int srcLane) {
    return __builtin_amdgcn_ds_bpermute(srcLane << 2, v);
}

__device__ __forceinline__ unsigned make_rowbase(int pg, int kvh, int idx) {
    // float offset of K row start: page*32768 + kvh*2048 + off*128 (fits u32)
    return ((unsigned)pg << 15) + ((unsigned)kvh << 11) + ((unsigned)(idx & 15) << 7);
}

__global__ __launch_bounds__(512, 1) void sparse_paged_attn_kernel(
    const float* __restrict__ q,         // [B,H,1,D]
    const float* __restrict__ kv,        // [pages,2,KVH,PS,D]
    const int*   __restrict__ indptr,    // [B+1]
    const int*   __restrict__ pindices,  // [total_pages]
    const int*   __restrict__ sind,      // [B,H,LMAX]
    const int*   __restrict__ snnz,      // [B,H]
    float*       __restrict__ out)       // [B,H,1,D]
{
    __shared__ float red_acc[WAVES_][D_];
    __shared__ float red_m[WAVES_];
    __shared__ float red_s[WAVES_];

    const int bh   = blockIdx.x;          // b*H + h
    const int b    = bh >> 5;
    const int h    = bh & 31;
    const int kvh  = h >> 2;              // groups = H/KVH = 4
    const int lane = threadIdx.x & 31;
    const int wave = threadIdx.x >> 5;
    const int hi   = lane >> 4;           // 0: lanes 0-15, 1: lanes 16-31
    const int row  = lane & 15;

    const int   nnz    = snnz[bh];
    const int   pstart = indptr[b];
    const float scale  = 0.08838834764831845f;   // 1/sqrt(128)

    // Preload q in WMMA B-matrix (4x16) layout, per 4-wide k-chunk c:
    //   VGPR0 = q[4c + hi*2], VGPR1 = q[4c + 1 + hi*2]
    const float* qp = q + (size_t)bh * D_;
    v2f bq[32];
#pragma unroll
    for (int c = 0; c < 32; ++c) {
        float2 t = *(const float2*)(qp + 4 * c + hi * 2);
        bq[c].x = t.x;
        bq[c].y = t.y;
    }

    float m_run = -1.0e30f;
    float s_run = 0.0f;
    float acc0 = 0.0f, acc1 = 0.0f, acc2 = 0.0f, acc3 = 0.0f;

    const int  nTiles  = (nnz + 15) >> 4;
    const int* sind_bh = sind + (size_t)bh * LMAX_;
    const float* vplane = kv + (KVH_ * PS_ * D_);  // +16384 floats: V plane

    int t = wave;
    if (t < nTiles) {
        // ---- pipeline prologue: row base for first tile ----
        int idx = sind_bh[t * 16 + row];
        int pg  = pindices[pstart + (idx >> 4)];
        unsigned rowbase = make_rowbase(pg, kvh, idx);

        for (; t < nTiles; t += WAVES_) {
            // ---- prefetch next tile's sparse index (critical latency path) ----
            const int tn    = t + WAVES_;
            const int ln    = (tn < nTiles) ? (tn * 16 + row) : (t * 16 + row);
            const int idx_n = sind_bh[ln];

            // ---- batch-issue all 32 K-chunk loads (WMMA A operands) ----
            // A (16x4) layout: lane m -> row m; lanes 0-15 get k0+0/1, 16-31 k0+2/3
            const v2f* ap = (const v2f*)(kv + rowbase + hi * 2);
            v2f ka[32];
#pragma unroll
            for (int cc = 0; cc < 32; ++cc) ka[cc] = ap[2 * cc];

            // ---- broadcast V row bases, batch-issue all 16 V row loads ----
            float4 vrow[16];
#pragma unroll
            for (int j = 0; j < 16; ++j) {
                unsigned vb = (unsigned)lanebcast_i((int)rowbase, j);
                vrow[j] = *(const float4*)(vplane + vb + 4 * lane);
            }

            // ---- second hop of next-tile pipeline (depends on idx_n load) ----
            const int pg_n = pindices[pstart + (idx_n >> 4)];

            // keep all 50 loads issued ahead of the WMMA chain: nothing may
            // be scheduled across this point
            __builtin_amdgcn_sched_barrier(0);

            // ---- QK: 32 chained V_WMMA_F32_16X16X4_F32 (exact fp32) ----
            v8f cacc = {0.f, 0.f, 0.f, 0.f, 0.f, 0.f, 0.f, 0.f};
#pragma unroll
            for (int cc = 0; cc < 32; ++cc) {
                cacc = __builtin_amdgcn_wmma_f32_16x16x4_f32(
                    false, ka[cc], false, bq[cc], (short)0, cacc, false, false);
            }
            // cacc[i] = qk of tile-row (hi*8 + i), replicated across columns

            // ---- masked scores + online softmax ----
            float sc[8];
#pragma unroll
            for (int i = 0; i < 8; ++i) {
                int j = hi * 8 + i;
                sc[i] = ((t * 16 + j) < nnz) ? cacc[i] * scale : -1.0e6f;
            }
            float tm = sc[0];
#pragma unroll
            for (int i = 1; i < 8; ++i) tm = fmaxf(tm, sc[i]);
            tm = fmaxf(tm, lanebcast_f(tm, lane ^ 16));     // cross-half max
            float m_new = fmaxf(m_run, tm);
            float fsc   = __expf(m_run - m_new);

            float pe[8];
            float tsum = 0.0f;
#pragma unroll
            for (int i = 0; i < 8; ++i) {
                pe[i] = __expf(sc[i] - m_new);   // masked -> exact 0 (underflow)
                tsum += pe[i];
            }
            tsum += lanebcast_f(tsum, lane ^ 16);
            s_run = s_run * fsc + tsum;
            m_run = m_new;
            acc0 *= fsc; acc1 *= fsc; acc2 *= fsc; acc3 *= fsc;

            // ---- PV: FMA from pre-loaded V rows, scalar p broadcast ----
#pragma unroll
            for (int j = 0; j < 16; ++j) {
                float pj = lanebcast_f(pe[j & 7], (j >> 3) << 4);
                acc0 = fmaf(pj, vrow[j].x, acc0);
                acc1 = fmaf(pj, vrow[j].y, acc1);
                acc2 = fmaf(pj, vrow[j].z, acc2);
                acc3 = fmaf(pj, vrow[j].w, acc3);
            }

            // ---- finish next-tile row base ----
            rowbase = make_rowbase(pg_n, kvh, idx_n);
        }
    }

    // ---- cross-wave flash combine through LDS ----
    if (lane == 0) {
        red_m[wave] = m_run;   // wave-uniform
        red_s[wave] = s_run;
    }
    red_acc[wave][4 * lane + 0] = acc0;
    red_acc[wave][4 * lane + 1] = acc1;
    red_acc[wave][4 * lane + 2] = acc2;
    red_acc[wave][4 * lane + 3] = acc3;
    __syncthreads();

    const int tid = threadIdx.x;
    if (tid < D_) {
        float M = -1.0e30f;
#pragma unroll
        for (int w = 0; w < WAVES_; ++w) M = fmaxf(M, red_m[w]);
        float num = 0.0f, den = 0.0f;
#pragma unroll
        for (int w = 0; w < WAVES_; ++w) {
            float f = __expf(red_m[w] - M);
            num = fmaf(f, red_acc[w][tid], num);
            den = fmaf(f, red_s[w], den);
        }
        out[(size_t)bh * D_ + tid] = num / den;
    }
}

extern "C" void kernel_launch(void* const* d_in, const int* in_sizes, int n_in,
                              void* d_out, int out_size, void* d_ws, size_t ws_size,
                              hipStream_t stream) {
    (void)in_sizes; (void)n_in; (void)d_ws; (void)ws_size; (void)out_size;
    const float* q        = (const float*)d_in[0];
    const float* kv       = (const float*)d_in[1];
    const int*   indptr   = (const int*)d_in[2];
    const int*   pindices = (const int*)d_in[3];
    const int*   sind     = (const int*)d_in[4];
    const int*   snnz     = (const int*)d_in[5];
    float*       out      = (float*)d_out;

    dim3 grid(B_ * H_);     // 256 blocks: one per (b,h)
    dim3 block(32 * WAVES_);
    sparse_paged_attn_kernel<<<grid, block, 0, stream>>>(
        q, kv, indptr, pindices, sind, snnz, out);
}